// DiffeqSolver_5385888989642
// MI455X (gfx1250) — compile-verified
//
#include <hip/hip_runtime.h>
#include <math.h>

// Neural-ODE RK4 solver for MI455X (gfx1250, wave32).
// Each (s,n) row is an independent 128-dim ODE trajectory: tile 16 rows per
// workgroup, keep state + both weight matrices in LDS (CDNA5: 320KB/WG),
// run all 99 RK4 steps in one kernel using V_WMMA_F32_16X16X4_F32.

typedef __attribute__((ext_vector_type(2))) float v2f;
typedef __attribute__((ext_vector_type(4))) float v4f;
typedef __attribute__((ext_vector_type(8))) float v8f;

#define LATS   128              // latent dim
#define HID    256              // hidden dim
#define TSTEPS 100              // time points (99 RK4 steps)
#define MTILE  16               // rows per workgroup (one WMMA M-tile)
#define NTHR   256              // 8 waves of 32
#define YSTR   (LATS + 4)       // padded LDS row stride (bank-conflict-free A loads)
#define HSTR   (HID + 4)

// gfx1250 has hardware V_TANH_F32 (TRANS pipe, co-executes with WMMA).
#if __has_builtin(__builtin_amdgcn_tanhf)
#define TANH_F32(x) __builtin_amdgcn_tanhf(x)
#else
#define TANH_F32(x) tanhf(x)
#endif

__launch_bounds__(NTHR, 1)
__global__ void node_rk4_kernel(const float* __restrict__ fp,
                                const float* __restrict__ tsg,
                                const float* __restrict__ W1g,
                                const float* __restrict__ b1g,
                                const float* __restrict__ W2g,
                                const float* __restrict__ b2g,
                                float* __restrict__ out)
{
    // ---- LDS: weights staged once, state resident for whole integration ----
    __shared__ float W1s[LATS * HID];      // 128 KB, row-major [k][n]
    __shared__ float W2s[HID * LATS];      // 128 KB, row-major [k][n]
    __shared__ float b1s[HID];
    __shared__ float b2s[LATS];
    __shared__ float tss[TSTEPS];
    __shared__ float y0s[MTILE * YSTR];    // step base state
    __shared__ float ycs[MTILE * YSTR];    // stage input state
    __shared__ float accs[MTILE * YSTR];   // k1 + 2k2 + 2k3 accumulator
    __shared__ float ks [MTILE * YSTR];    // current stage k = f(y)
    __shared__ float hs [MTILE * HSTR];    // tanh hidden activations

    const int tid  = threadIdx.x;
    const int wg   = blockIdx.x;
    const int wave = tid >> 5;
    const int lane = tid & 31;
    // WMMA f32 16x16x4 fragment geometry (ISA 7.12.2):
    //   A (16x4): lanes 0-15 carry M=lane with K={0,1}; lanes 16-31 K={2,3}
    //   B (4x16): lanes 0-15 carry N=lane with K={0,1}; lanes 16-31 K={2,3}
    //   C/D (16x16): lane%16 = N; vgpr i = M (lanes<16: M=i, lanes>=16: M=i+8)
    const int rA  = lane & 15;
    const int kHi = (lane >> 4) << 1;      // 0 or 2
    const int nB  = lane & 15;
    const int mB  = (lane >> 4) * 8;

    // ---- stage weights/biases/times into LDS ----
    for (int i = tid * 4; i < LATS * HID; i += NTHR * 4) {
        *(v4f*)&W1s[i] = *(const v4f*)&W1g[i];
        *(v4f*)&W2s[i] = *(const v4f*)&W2g[i];
    }
    if (tid < HID)    b1s[tid] = b1g[tid];
    if (tid < LATS)   b2s[tid] = b2g[tid];
    if (tid < TSTEPS) tss[tid] = tsg[tid];

    // ---- load initial state; emit t=0 output (streaming stores) ----
    const int  rOut = tid >> 4;            // 0..15
    const int  cOut = (tid & 15) * 8;      // 0,8,...,120
    const long rowG = (long)wg * MTILE + rOut;
    const int  eB   = rOut * YSTR + cOut;  // this thread's 8 state elements
    {
        v4f a = *(const v4f*)&fp[rowG * LATS + cOut];
        v4f b = *(const v4f*)&fp[rowG * LATS + cOut + 4];
        *(v4f*)&y0s[eB]     = a;
        *(v4f*)&y0s[eB + 4] = b;
        v4f* o = (v4f*)&out[(rowG * TSTEPS) * LATS + cOut];
        __builtin_nontemporal_store(a, o);
        __builtin_nontemporal_store(b, o + 1);
    }
    __syncthreads();

    // ---- k = f(yin) = tanh(yin@W1+b1)@W2+b2, result into ks ----
    auto eval_f = [&](const float* yin) {
        // GEMM1: (16x128)@(128x256) -> hs ; 2 N-tiles per wave
        const int n0 = wave * 32;
        v8f c0, c1;
        {
            float bi0 = b1s[n0 + nB], bi1 = b1s[n0 + 16 + nB];
            #pragma unroll
            for (int i = 0; i < 8; i++) { c0[i] = bi0; c1[i] = bi1; }
        }
        #pragma unroll
        for (int kk = 0; kk < LATS; kk += 4) {
            const int kr = kk + kHi;
            v2f a = *(const v2f*)&yin[rA * YSTR + kr];
            v2f b0, b1v;
            b0.x  = W1s[kr * HID + n0 + nB];
            b0.y  = W1s[(kr + 1) * HID + n0 + nB];
            b1v.x = W1s[kr * HID + n0 + 16 + nB];
            b1v.y = W1s[(kr + 1) * HID + n0 + 16 + nB];
            c0 = __builtin_amdgcn_wmma_f32_16x16x4_f32(false, a, false, b0,  (short)0, c0, false, false);
            c1 = __builtin_amdgcn_wmma_f32_16x16x4_f32(false, a, false, b1v, (short)0, c1, false, false);
        }
        #pragma unroll
        for (int i = 0; i < 8; i++) {
            hs[(mB + i) * HSTR + n0 + nB]      = TANH_F32(c0[i]);
            hs[(mB + i) * HSTR + n0 + 16 + nB] = TANH_F32(c1[i]);
        }
        __syncthreads();

        // GEMM2: (16x256)@(256x128) -> ks ; 1 N-tile per wave, 2 K-chains
        const int n2 = wave * 16;
        v8f d0, d1;
        {
            float bi2 = b2s[n2 + nB];
            #pragma unroll
            for (int i = 0; i < 8; i++) { d0[i] = bi2; d1[i] = 0.f; }
        }
        #pragma unroll
        for (int kk = 0; kk < HID; kk += 8) {
            const int kr0 = kk + kHi, kr1 = kk + 4 + kHi;
            v2f a0 = *(const v2f*)&hs[rA * HSTR + kr0];
            v2f a1 = *(const v2f*)&hs[rA * HSTR + kr1];
            v2f b0, b1v;
            b0.x  = W2s[kr0 * LATS + n2 + nB];
            b0.y  = W2s[(kr0 + 1) * LATS + n2 + nB];
            b1v.x = W2s[kr1 * LATS + n2 + nB];
            b1v.y = W2s[(kr1 + 1) * LATS + n2 + nB];
            d0 = __builtin_amdgcn_wmma_f32_16x16x4_f32(false, a0, false, b0,  (short)0, d0, false, false);
            d1 = __builtin_amdgcn_wmma_f32_16x16x4_f32(false, a1, false, b1v, (short)0, d1, false, false);
        }
        #pragma unroll
        for (int i = 0; i < 8; i++)
            ks[(mB + i) * YSTR + n2 + nB] = d0[i] + d1[i];
        __syncthreads();
    };

    // ---- RK4 time loop (sequential; keep body un-unrolled) ----
    #pragma unroll 1
    for (int step = 0; step < TSTEPS - 1; ++step) {
        const float dt = tss[step + 1] - tss[step];
        const float hf = 0.5f * dt;

        eval_f(y0s);                                   // k1
        #pragma unroll
        for (int j = 0; j < 8; j++) { float k = ks[eB+j]; accs[eB+j] = k;        ycs[eB+j] = y0s[eB+j] + hf * k; }
        __syncthreads();

        eval_f(ycs);                                   // k2
        #pragma unroll
        for (int j = 0; j < 8; j++) { float k = ks[eB+j]; accs[eB+j] += 2.f * k; ycs[eB+j] = y0s[eB+j] + hf * k; }
        __syncthreads();

        eval_f(ycs);                                   // k3
        #pragma unroll
        for (int j = 0; j < 8; j++) { float k = ks[eB+j]; accs[eB+j] += 2.f * k; ycs[eB+j] = y0s[eB+j] + dt * k; }
        __syncthreads();

        eval_f(ycs);                                   // k4
        const float s6 = dt * (1.0f / 6.0f);
        #pragma unroll
        for (int j = 0; j < 8; j++) { y0s[eB+j] += s6 * (accs[eB+j] + ks[eB+j]); }
        __syncthreads();

        // emit y(t_{step+1}) : out[(row*T + t)*L + c], streaming store
        v4f a = *(const v4f*)&y0s[eB];
        v4f b = *(const v4f*)&y0s[eB + 4];
        v4f* o = (v4f*)&out[(rowG * TSTEPS + (step + 1)) * LATS + cOut];
        __builtin_nontemporal_store(a, o);
        __builtin_nontemporal_store(b, o + 1);
    }
}

extern "C" void kernel_launch(void* const* d_in, const int* in_sizes, int n_in,
                              void* d_out, int out_size, void* d_ws, size_t ws_size,
                              hipStream_t stream) {
    const float* fp = (const float*)d_in[0];   // first_point (4,1024,128)
    const float* ts = (const float*)d_in[1];   // time_steps (100,)
    const float* W1 = (const float*)d_in[2];   // (128,256)
    const float* b1 = (const float*)d_in[3];   // (256,)
    const float* W2 = (const float*)d_in[4];   // (256,128)
    const float* b2 = (const float*)d_in[5];   // (128,)
    float* out = (float*)d_out;                // (4,1024,100,128)

    const int rows = in_sizes[0] / LATS;       // 4096 independent trajectories
    dim3 grid(rows / MTILE);                   // 256 workgroups
    node_rk4_kernel<<<grid, NTHR, 0, stream>>>(fp, ts, W1, b1, W2, b2, out);
}